// MANO_Model_61443802136609
// MI455X (gfx1250) — compile-verified
//
#include <hip/hip_runtime.h>
#include <hip/hip_bf16.h>
#include <math.h>

#define B_BATCH 8192
#define V_NUM   778
#define NJOINT  16
#define KP      148      // 135 pose_feat + 10 beta + 3 zero pad (multiple of 4)
#define KSTEPS  37       // KP/4
#define NCOL    2334     // 778*3
#define NT_TILES 148     // N padded to 148 tiles of 16 -> 2368 cols (multiple of 64)
#define VHS     2368     // Vh row stride
#define MG      (B_BATCH/32)   // 256 M-groups (32 rows each)
#define NG      (NT_TILES/4)   // 37 N-groups (64 cols each)

typedef __attribute__((ext_vector_type(2))) float v2f;
typedef __attribute__((ext_vector_type(8))) float v8f;

__constant__ int c_ftip[5]   = {745, 333, 444, 555, 672};
__constant__ int c_remap[21] = {0,16,15,14,13,17,3,2,1,18,6,5,4,19,12,11,10,20,9,8,7};

// ---------------------------------------------------------------- helpers
__device__ inline void rodrigues3(const float* rv, float* R) {
    float x = rv[0], y = rv[1], z = rv[2];
    float ang = sqrtf(x*x + y*y + z*z) + 1e-8f;
    float inv = 1.0f / ang;
    float kx = x*inv, ky = y*inv, kz = z*inv;
    float s = sinf(ang), c = cosf(ang);
    float C = 1.0f - c;
    R[0] = c + C*kx*kx;     R[1] = -s*kz + C*kx*ky; R[2] =  s*ky + C*kx*kz;
    R[3] =  s*kz + C*ky*kx; R[4] = c + C*ky*ky;     R[5] = -s*kx + C*ky*kz;
    R[6] = -s*ky + C*kz*kx; R[7] =  s*kx + C*kz*ky; R[8] = c + C*kz*kz;
}

__device__ inline void mm3(const float* A, const float* Bm, float* Cm) {
    #pragma unroll
    for (int i = 0; i < 3; i++)
        #pragma unroll
        for (int j = 0; j < 3; j++)
            Cm[i*3+j] = A[i*3+0]*Bm[0*3+j] + A[i*3+1]*Bm[1*3+j] + A[i*3+2]*Bm[2*3+j];
}

__device__ inline void jrest_calc(const float* JS, const float* JT, const float* bt,
                                  int j, float* out) {
    #pragma unroll
    for (int t = 0; t < 3; t++) {
        float acc = JT[j*3+t];
        const float* row = JS + (j*3+t)*10;
        #pragma unroll
        for (int l = 0; l < 10; l++) acc += row[l]*bt[l];
        out[t] = acc;
    }
}

__device__ inline void write_gp(float* Gp, float* pj, int b, int j,
                                const float* R, const float* t, const float* Jr) {
    float* g = Gp + ((size_t)b*NJOINT + j)*12;
    #pragma unroll
    for (int p = 0; p < 3; p++) {
        g[p*4+0] = R[p*3+0]; g[p*4+1] = R[p*3+1]; g[p*4+2] = R[p*3+2];
        g[p*4+3] = t[p] - (R[p*3+0]*Jr[0] + R[p*3+1]*Jr[1] + R[p*3+2]*Jr[2]);
    }
    float* pp = pj + ((size_t)b*NJOINT + j)*3;
    pp[0] = t[0]; pp[1] = t[1]; pp[2] = t[2];
}

// ------------------------------------------- kernel 0a: JS = J*S, JT = J*T_bar
__global__ void precompute_js(const float* __restrict__ Jm, const float* __restrict__ S,
                              const float* __restrict__ T_bar,
                              float* __restrict__ JS, float* __restrict__ JT) {
    int idx = blockIdx.x*blockDim.x + threadIdx.x;
    if (idx < 480) {
        int j = idx/30, rem = idx%30, t = rem/10, l = rem%10;
        float acc = 0.f;
        for (int v = 0; v < V_NUM; v++) acc += Jm[j*V_NUM+v]*S[v*30 + t*10 + l];
        JS[idx] = acc;
    } else if (idx < 528) {
        int jt = idx - 480, j = jt/3, t = jt%3;
        float acc = 0.f;
        for (int v = 0; v < V_NUM; v++) acc += Jm[j*V_NUM+v]*T_bar[v*3+t];
        JT[jt] = acc;
    }
}

// ------------------------------------------- kernel 0b: B fragments
// Combined matrix rows 0..134 = P, 135..144 = S^T, 145..147 = 0, cols padded to 2368.
// Stored in WMMA 4x16 f32 fragment order:
//   Bfrag[((nt*KSTEPS + kstep)*32 + lane)*2 + e] holds element
//   (K = kstep*4 + (lane>>4)*2 + e, col = nt*16 + (lane&15))
// so the GEMM loads one b64 per lane per fragment, fully coalesced.
__global__ void build_bfrag(const float* __restrict__ P, const float* __restrict__ S,
                            float* __restrict__ Bf) {
    int idx = blockIdx.x*blockDim.x + threadIdx.x;
    if (idx >= NT_TILES*KSTEPS*64) return;
    int e    = idx & 1;
    int lane = (idx >> 1) & 31;
    int frag = idx >> 6;            // nt*KSTEPS + kstep
    int ks   = frag % KSTEPS;
    int nt   = frag / KSTEPS;
    int K    = ks*4 + ((lane >> 4) << 1) + e;
    int col  = nt*16 + (lane & 15);
    float val = 0.f;
    if (col < NCOL && K < 145) {
        if (K < 135) val = P[K*NCOL + col];
        else         val = S[(col/3)*30 + (col%3)*10 + (K-135)];
    }
    Bf[idx] = val;
}

// ------------------------------------------- kernel 1: per-batch prep
__global__ __launch_bounds__(256)
void batch_prep(const float* __restrict__ beta, const float* __restrict__ pose,
                const float* __restrict__ JS, const float* __restrict__ JT,
                float* __restrict__ feat, float* __restrict__ Gp,
                float* __restrict__ pjoints) {
    int b = blockIdx.x*blockDim.x + threadIdx.x;
    if (b >= B_BATCH) return;

    float bt[10];
    #pragma unroll
    for (int l = 0; l < 10; l++) bt[l] = beta[b*10 + l];

    float* fb = feat + (size_t)b*KP;
    #pragma unroll
    for (int l = 0; l < 10; l++) fb[135 + l] = bt[l];
    fb[145] = 0.f; fb[146] = 0.f; fb[147] = 0.f;

    float R0[9], J0[3];
    rodrigues3(pose + (size_t)b*48, R0);
    jrest_calc(JS, JT, bt, 0, J0);
    write_gp(Gp, pjoints, b, 0, R0, J0, J0);   // root: t = rel = J_rest[0]

    // five 3-joint chains off the root: joints 1..3, 4..6, 7..9, 10..12, 13..15
    for (int ch = 0; ch < 5; ch++) {
        float GR[9], Gt[3], pJ[3];
        #pragma unroll
        for (int i = 0; i < 9; i++) GR[i] = R0[i];
        #pragma unroll
        for (int i = 0; i < 3; i++) { Gt[i] = J0[i]; pJ[i] = J0[i]; }
        for (int k = 0; k < 3; k++) {
            int j = 1 + ch*3 + k;
            float Rl[9];
            rodrigues3(pose + (size_t)b*48 + j*3, Rl);
            #pragma unroll
            for (int i = 0; i < 9; i++)
                fb[(j-1)*9 + i] = Rl[i] - ((i == 0 || i == 4 || i == 8) ? 1.f : 0.f);
            float Jr[3];
            jrest_calc(JS, JT, bt, j, Jr);
            float rel[3] = {Jr[0]-pJ[0], Jr[1]-pJ[1], Jr[2]-pJ[2]};
            float nR[9], nt[3];
            mm3(GR, Rl, nR);
            #pragma unroll
            for (int p = 0; p < 3; p++)
                nt[p] = GR[p*3+0]*rel[0] + GR[p*3+1]*rel[1] + GR[p*3+2]*rel[2] + Gt[p];
            write_gp(Gp, pjoints, b, j, nR, nt, Jr);
            #pragma unroll
            for (int i = 0; i < 9; i++) GR[i] = nR[i];
            #pragma unroll
            for (int i = 0; i < 3; i++) { Gt[i] = nt[i]; pJ[i] = Jr[i]; }
        }
    }
}

// ------------------------------------------- kernel 2: WMMA f32 GEMM
// Vh(B x 2368) = T_bar_broadcast + feat(B x 148) @ Bcomb(148 x 2368)
// Register-blocked 32(M) x 64(N) per wave: 2 A-frag + 4 B-frag -> 8 WMMAs / K-step.
__global__ __launch_bounds__(256)
void gemm_wmma(const float* __restrict__ feat, const float* __restrict__ Bf,
               const float* __restrict__ T_bar, float* __restrict__ Vh) {
    int lane = threadIdx.x & 31;
    int wave = threadIdx.x >> 5;
    int tile = blockIdx.x*8 + wave;           // 8 waves per block
    int mg   = tile / NG;                     // M-group: 32 batch rows
    int ng   = tile % NG;                     // N-group: 4 column tiles (64 cols)
    if (mg >= MG) return;                     // wave-uniform

    int half = lane >> 4;                     // 0: K=k,k+1   1: K=k+2,k+3
    int l15  = lane & 15;

    int colv[4];
    v8f acc[2][4];
    #pragma unroll
    for (int t = 0; t < 4; t++) {
        colv[t] = (ng*4 + t)*16 + l15;
        float tb = (colv[t] < NCOL) ? T_bar[colv[t]] : 0.f;  // C init = T_bar
        #pragma unroll
        for (int r = 0; r < 8; r++) { acc[0][t][r] = tb; acc[1][t][r] = tb; }
    }

    const float* a0p = feat + (size_t)(mg*32 + l15)*KP;
    const float* a1p = a0p + (size_t)16*KP;
    for (int ks = 0; ks < KSTEPS; ks++) {
        int ka = ks*4 + half*2;                       // even -> 8B aligned
        v2f a0 = *(const v2f*)(a0p + ka);
        v2f a1 = *(const v2f*)(a1p + ka);
        if (ks + 4 < KSTEPS)
            __builtin_prefetch(Bf + ((size_t)((ng*4)*KSTEPS + ks + 4)*32 + lane)*2, 0, 1);
        #pragma unroll
        for (int t = 0; t < 4; t++) {
            v2f bb = *(const v2f*)(Bf + ((size_t)((ng*4 + t)*KSTEPS + ks)*32 + lane)*2);
            acc[0][t] = __builtin_amdgcn_wmma_f32_16x16x4_f32(
                            false, a0, false, bb, (short)0, acc[0][t], false, false);
            acc[1][t] = __builtin_amdgcn_wmma_f32_16x16x4_f32(
                            false, a1, false, bb, (short)0, acc[1][t], false, false);
        }
    }

    // C/D layout: VGPR r -> M = r (lanes 0-15) / M = r+8 (lanes 16-31)
    #pragma unroll
    for (int m = 0; m < 2; m++) {
        int rbase = mg*32 + m*16 + half*8;
        #pragma unroll
        for (int t = 0; t < 4; t++)
            #pragma unroll
            for (int r = 0; r < 8; r++)
                Vh[(size_t)(rbase + r)*VHS + colv[t]] = acc[m][t][r];
    }
}

// ------------------------------------------- kernel 3: LBS skinning -> mesh
__global__ __launch_bounds__(256)
void skin(const float* __restrict__ Vh, const float* __restrict__ W,
          const float* __restrict__ Gp, float* __restrict__ mesh) {
    int b = blockIdx.x;
    __shared__ float G[NJOINT*12];
    for (int i = threadIdx.x; i < NJOINT*12; i += blockDim.x)
        G[i] = Gp[(size_t)b*NJOINT*12 + i];
    __syncthreads();

    const float* vb = Vh + (size_t)b*VHS;
    for (int v = threadIdx.x; v < V_NUM; v += blockDim.x) {
        float x = vb[v*3+0], y = vb[v*3+1], z = vb[v*3+2];
        float L[12];
        #pragma unroll
        for (int i = 0; i < 12; i++) L[i] = 0.f;
        #pragma unroll
        for (int j = 0; j < NJOINT; j++) {
            float w = W[v*NJOINT + j];
            #pragma unroll
            for (int i = 0; i < 12; i++) L[i] += w*G[j*12 + i];
        }
        float* o = mesh + ((size_t)b*V_NUM + v)*3;
        o[0] = L[0]*x + L[1]*y + L[2]*z  + L[3];
        o[1] = L[4]*x + L[5]*y + L[6]*z  + L[7];
        o[2] = L[8]*x + L[9]*y + L[10]*z + L[11];
    }
}

// ------------------------------------------- kernel 4: joints gather + REMAP
__global__ void joints_out(const float* __restrict__ pjoints,
                           const float* __restrict__ mesh,
                           float* __restrict__ joints) {
    int idx = blockIdx.x*blockDim.x + threadIdx.x;
    if (idx >= B_BATCH*21) return;
    int b = idx / 21, jo = idx % 21;
    int src = c_remap[jo];
    const float* s;
    if (src < NJOINT) s = pjoints + ((size_t)b*NJOINT + src)*3;
    else              s = mesh + ((size_t)b*V_NUM + c_ftip[src - NJOINT])*3;
    float* o = joints + ((size_t)b*21 + jo)*3;
    o[0] = s[0]; o[1] = s[1]; o[2] = s[2];
}

// ---------------------------------------------------------------- launch
extern "C" void kernel_launch(void* const* d_in, const int* in_sizes, int n_in,
                              void* d_out, int out_size, void* d_ws, size_t ws_size,
                              hipStream_t stream) {
    const float* beta  = (const float*)d_in[0];
    const float* pose  = (const float*)d_in[1];
    const float* T_bar = (const float*)d_in[2];
    const float* S     = (const float*)d_in[3];
    const float* P     = (const float*)d_in[4];
    const float* Jm    = (const float*)d_in[5];
    const float* W     = (const float*)d_in[6];
    // d_in[7] = parents (hardcoded MANO tree)

    float* ws = (float*)d_ws;
    float* JS    = ws + 0;                         // 480
    float* JT    = ws + 512;                       // 48
    float* Bfrag = ws + 768;                       // 148*37*64 = 350464
    float* feat  = ws + 351232;                    // 8192*148  = 1212416
    float* Gp    = ws + 351232 + 1212416;          // 8192*192  = 1572864
    float* pj    = ws + 1563648 + 1572864;         // 8192*48   = 393216
    float* Vh    = ws + 3136512 + 393216;          // 8192*2368 = 19398656

    float* mesh   = (float*)d_out;
    float* joints = mesh + (size_t)B_BATCH*V_NUM*3;

    precompute_js<<<3, 256, 0, stream>>>(Jm, S, T_bar, JS, JT);
    build_bfrag<<<(NT_TILES*KSTEPS*64 + 255)/256, 256, 0, stream>>>(P, S, Bfrag);
    batch_prep<<<B_BATCH/256, 256, 0, stream>>>(beta, pose, JS, JT, feat, Gp, pj);
    {
        int waves  = MG * NG;                      // 256*37 = 9472
        int blocks = (waves + 7) / 8;              // 1184, exact
        gemm_wmma<<<blocks, 256, 0, stream>>>(feat, Bfrag, T_bar, Vh);
    }
    skin<<<B_BATCH, 256, 0, stream>>>(Vh, W, Gp, mesh);
    joints_out<<<(B_BATCH*21 + 255)/256, 256, 0, stream>>>(pj, mesh, joints);
}